// DynamicFilter_79448305041858
// MI455X (gfx1250) — compile-verified
//
#include <hip/hip_runtime.h>

// ---------------------------------------------------------------------------
// DynamicFilter on MI455X (gfx1250).
//
// Key observation: the reference applies softmax over an axis of size 1, so
// the generated dynamic filter is identically 1.0.  The op reduces to a 3x3
// reflect-padded box SUM per channel:  out = A * X * A^T  with A the constant
// 56x56 tridiagonal "reflect" matrix (A[i][j]=1 for |i-j|<=1, A[0][1]=2,
// A[55][54]=2).  This is memory-bound: 77MB in + 77MB out -> ~6.6us at
// 23.3 TB/s (and the 154MB working set fits the 192MB L2 for timed replays,
// so default RT cache policy is used, no NT hints).  The separable stencil
// runs through V_WMMA_F32_16X16X4_F32 (full fp32; multiplies by {0,1,2} are
// exact) with the banded A synthesized in registers, and X staged via CDNA5
// async global->LDS b128 copies waited with s_wait_asynccnt.
//
// One wave32 per (n,c) channel; 16*384 = 6144 workgroups, 30.4KB LDS each
// (~10 concurrent channels per WGP; ~125KB of async loads in flight per WGP).
// ---------------------------------------------------------------------------

typedef float v2f __attribute__((ext_vector_type(2)));
typedef float v4f __attribute__((ext_vector_type(4)));
typedef float v8f __attribute__((ext_vector_type(8)));
typedef int   v4i __attribute__((ext_vector_type(4)));

typedef __attribute__((address_space(1))) v4i* gv4i_p; // global 16B chunk
typedef __attribute__((address_space(3))) v4i* lv4i_p; // LDS 16B chunk

#define HW 56          // image height == width
#define LS 68          // LDS row stride in floats (272B/row -> conflict-free strided b64)
#define NPIX (HW * HW) // 3136 floats per channel

#if __has_builtin(__builtin_amdgcn_global_load_async_to_lds_b128)
#define HAVE_ASYNC_LDS 1
#endif

// Banded reflect matrix entry A[i][j] (zero outside [0,56)).
__device__ __forceinline__ float Aval(int i, int j) {
  int d = i - j;
  int band = ((unsigned)i < (unsigned)HW) & ((unsigned)j < (unsigned)HW) &
             (d >= -1) & (d <= 1);
  int dbl = ((i == 0) & (j == 1)) | ((i == HW - 1) & (j == HW - 2));
  return band ? (dbl ? 2.0f : 1.0f) : 0.0f;
}

__global__ __launch_bounds__(32) void DynamicFilter_79448305041858_kernel(
    const float* __restrict__ x, float* __restrict__ out) {
  // Two LDS planes: input tile X and intermediate U = X * A^T.
  __shared__ float lX[HW * LS]; // 15232 B
  __shared__ float lU[HW * LS]; // 15232 B

  const int lane = threadIdx.x & 31;
  const int half = lane >> 4;  // 0: lanes 0-15, 1: lanes 16-31
  const int l16  = lane & 15;

  const long long ch = blockIdx.x;
  const float* __restrict__ src = x + ch * NPIX;
  float* __restrict__ dst = out + ch * NPIX;

  // ---- Stage 0: channel image -> LDS (row stride 56 -> 68) ----------------
  // 784 groups of 4 floats; 56 % 4 == 0 so groups never cross a row.
  for (int g = lane; g < NPIX / 4; g += 32) {
    int row = g / 14;
    int col = (g - row * 14) * 4;
#if defined(HAVE_ASYNC_LDS)
    __builtin_amdgcn_global_load_async_to_lds_b128(
        (gv4i_p)(src + row * HW + col),
        (lv4i_p)(&lX[row * LS + col]),
        /*offset=*/0, /*cpol=*/0);
#else
    *(v4f*)(&lX[row * LS + col]) = *(const v4f*)(src + row * HW + col);
#endif
  }
#if defined(HAVE_ASYNC_LDS)
#if __has_builtin(__builtin_amdgcn_s_wait_asynccnt)
  __builtin_amdgcn_s_wait_asynccnt(0);
#else
  asm volatile("s_wait_asynccnt 0" ::: "memory");
#endif
  // Compiler-level fence: ASYNCcnt is not modeled as a dependence of the
  // ds_loads below; keep every lX read after the wait.
  asm volatile("" ::: "memory");
#endif
  // Single wave owns this LDS; same-wave DS ops are in order -> no barrier.

  // ---- Stage 1: U = X * A^T  (horizontal 3-sums) --------------------------
  // nt outermost: the A^T B-operand (register-synthesized) for each K-step is
  // built once and reused across all four mt tiles.  Band structure limits
  // the K-range per column tile: k in [16*nt-4, min(16*nt+16, 52)].
#pragma unroll
  for (int nt = 0; nt < 4; ++nt) {
    const int ks = (nt * 16 - 4) > 0 ? (nt * 16 - 4) : 0;
    const int ke = (nt * 16 + 16) < 52 ? (nt * 16 + 16) : 52;
    // B-operand layout (4x16): VGPR0/1 = (K = k+2*half +0/+1, N = l16).
    v2f bArr[6];
#pragma unroll
    for (int k = ks; k <= ke; k += 4) {
      const int cc = k + 2 * half;  // true K index, <= 54
      const int n  = nt * 16 + l16;
      bArr[(k - ks) >> 2].x = Aval(n, cc);      // A^T[cc][n]
      bArr[(k - ks) >> 2].y = Aval(n, cc + 1);
    }
#pragma unroll
    for (int mt = 0; mt < 4; ++mt) {
      v8f acc = {};
#pragma unroll
      for (int k = ks; k <= ke; k += 4) {
        // A-operand layout (16x4 f32): VGPR0/1 = (M = l16, K = k+2*half +0/+1).
        int row = mt * 16 + l16;                // up to 63 for mt == 3
        row = row > (HW - 1) ? (HW - 1) : row;  // clamp: rows >= 56 dropped at store
        const int cc = k + 2 * half;            // 8B-aligned column
        v2f a = *(const v2f*)(&lX[row * LS + cc]);
        acc = __builtin_amdgcn_wmma_f32_16x16x4_f32(
            false, a, false, bArr[(k - ks) >> 2], (short)0, acc, false, false);
      }
      // C/D tile -> LDS: VGPR r holds U[mt*16 + r + 8*half][nt*16 + l16].
#pragma unroll
      for (int r = 0; r < 8; ++r) {
        const int row = mt * 16 + r + 8 * half;
        if (row < HW) lU[row * LS + nt * 16 + l16] = acc[r];
      }
    }
  }

  // ---- Stage 2: Z = A * U  (vertical 3-sums), store to global -------------
  // mt outermost: the banded-A A-operand per K-step is built once and reused
  // across all four nt tiles.
#pragma unroll
  for (int mt = 0; mt < 4; ++mt) {
    const int ks = (mt * 16 - 4) > 0 ? (mt * 16 - 4) : 0;
    const int ke = (mt * 16 + 16) < 52 ? (mt * 16 + 16) : 52;
    v2f aArr[6];
#pragma unroll
    for (int k = ks; k <= ke; k += 4) {
      const int kk = k + 2 * half;  // <= 54
      aArr[(k - ks) >> 2].x = Aval(mt * 16 + l16, kk);
      aArr[(k - ks) >> 2].y = Aval(mt * 16 + l16, kk + 1);
    }
#pragma unroll
    for (int nt = 0; nt < 4; ++nt) {
      v8f acc = {};
#pragma unroll
      for (int k = ks; k <= ke; k += 4) {
        const int kk = k + 2 * half;
        // B-operand = U rows kk, kk+1 (cols nt*16+l16; cols <= 63 valid).
        const int n = nt * 16 + l16;
        v2f b;
        b.x = lU[kk * LS + n];
        b.y = lU[(kk + 1) * LS + n];
        acc = __builtin_amdgcn_wmma_f32_16x16x4_f32(
            false, aArr[(k - ks) >> 2], false, b, (short)0, acc, false, false);
      }
      // Store C/D tile: two contiguous 64B row segments per VGPR (coalesced).
#pragma unroll
      for (int r = 0; r < 8; ++r) {
        const int row = mt * 16 + r + 8 * half;
        const int col = nt * 16 + l16;
        if (row < HW && col < HW) dst[row * HW + col] = acc[r];
      }
    }
  }
}

extern "C" void kernel_launch(void* const* d_in, const int* in_sizes, int n_in,
                              void* d_out, int out_size, void* d_ws, size_t ws_size,
                              hipStream_t stream) {
  (void)n_in; (void)d_ws; (void)ws_size; (void)out_size;
  const float* x = (const float*)d_in[0];
  float* out = (float*)d_out;
  // All other inputs (w_conv, w_gate, BN params) are mathematically unused:
  // softmax over a singleton axis makes the dynamic filter identically 1.
  const int nch = in_sizes[0] / NPIX; // 16 * 384 = 6144 channel images
  dim3 grid(nch), block(32);
  DynamicFilter_79448305041858_kernel<<<grid, block, 0, stream>>>(x, out);
}